// ScaledDotProductAttention_755914244304
// MI455X (gfx1250) — compile-verified
//
#include <hip/hip_runtime.h>

typedef float v2f __attribute__((ext_vector_type(2)));
typedef float v8f __attribute__((ext_vector_type(8)));

constexpr int B_ = 4, H_ = 16, S_ = 1024, D_ = 64;
constexpr float INV_TEMP = 0.125f;   // 1/8

#define WMMA_F32_16x16x4(a, b, c) \
  __builtin_amdgcn_wmma_f32_16x16x4_f32(false, (a), false, (b), (short)0, (c), false, false)

// XOR bank swizzle: physical col = col ^ (4*row). Keeps aligned 4-float groups
// contiguous, removes the 16-way conflict on phase-3 same-column row reads.
__device__ __forceinline__ int swz(int row, int col) {
  return row * S_ + (col ^ ((row & 15) * 4));
}

// ---------------------------------------------------------------------------
// Fully fused attention. One workgroup (8 wave32) owns a 16-row query block:
//   Phase 1: scores = QK^T/8 + mask*gamma (+fill)  -> LDS (WMMA)
//   Phase 2: row softmax in LDS, single coalesced write of attn to HBM
//   Phase 3: out = scores @ V from LDS (WMMA, split-K across wave pairs)
// attn (256 MB) touches HBM exactly once.
// ---------------------------------------------------------------------------
__global__ __launch_bounds__(256) void fused_attention_kernel(
    const float* __restrict__ q, const float* __restrict__ k,
    const float* __restrict__ v, const float* __restrict__ mask,
    const int* __restrict__ amask, const float* __restrict__ gamma_p,
    float* __restrict__ out, float* __restrict__ attn)
{
  __shared__ float sc[16 * S_];   // 64 KB score stripe (of 320 KB/WGP)

  const int lane = threadIdx.x & 31;
  const int wave = threadIdx.x >> 5;
  const int m    = lane & 15;     // fragment row/col index
  const int half = lane >> 4;     // 0: K%4 in {0,1}; 1: K%4 in {2,3}

  const int rb    = blockIdx.x % (S_ / 16);   // 16-row query block
  const int bh    = blockIdx.x / (S_ / 16);
  const int b_idx = bh / H_;

  const float gamma = gamma_p[0];
  const float* __restrict__ qb = q + (size_t)bh * S_ * D_;
  const float* __restrict__ kb = k + (size_t)bh * S_ * D_;
  const int qrow = rb * 16 + m;

  // ---------------- Phase 1: scores -> LDS ----------------
  // 64 col-tiles over 8 waves; each wave: 2 groups of 4 tiles (A reused).
  for (int g = 0; g < 2; ++g) {
    const int ct0 = wave * 8 + g * 4;
    v8f acc[4] = {v8f{}, v8f{}, v8f{}, v8f{}};
#pragma unroll
    for (int kk = 0; kk < D_; kk += 4) {
      const int ko = kk + half * 2;
      v2f a;
      a.x = qb[qrow * D_ + ko];
      a.y = qb[qrow * D_ + ko + 1];
#pragma unroll
      for (int nt = 0; nt < 4; ++nt) {
        const int c = (ct0 + nt) * 16 + m;
        v2f bf;
        bf.x = kb[c * D_ + ko];
        bf.y = kb[c * D_ + ko + 1];
        acc[nt] = WMMA_F32_16x16x4(a, bf, acc[nt]);
      }
    }
    const float* __restrict__ mb = mask  + (size_t)b_idx * S_ * S_;
    const int*   __restrict__ ab = amask + (size_t)b_idx * S_ * S_;
#pragma unroll
    for (int r = 0; r < 8; ++r) {
      const int mm   = r + half * 8;        // local row 0..15
      const int grow = rb * 16 + mm;        // global row
#pragma unroll
      for (int nt = 0; nt < 4; ++nt) {
        const int nn = (ct0 + nt) * 16 + m;
        const size_t gidx = (size_t)grow * S_ + nn;
        float val = acc[nt][r] * INV_TEMP + mb[gidx] * gamma;
        if (ab[gidx] == 0) val = -1e-9f;
        sc[swz(mm, nn)] = val;
      }
    }
  }
  __syncthreads();

  // ---------------- Phase 2: softmax in LDS + attn writeback ----------------
  // 16 rows over 8 waves -> 2 rows/wave; 32 lanes * 8 float4 = 1024 elems.
  float* __restrict__ attnB = attn + (size_t)bh * S_ * S_;
#pragma unroll
  for (int i = 0; i < 2; ++i) {
    const int r  = wave * 2 + i;
    const int sx = r * 4;                   // swizzle constant for this row
    float4 x[8];
    float mx = -3.402823466e38f;
#pragma unroll
    for (int c2 = 0; c2 < 8; ++c2) {
      const int j   = lane + c2 * 32;       // logical float4-group index
      const int col = (4 * j) ^ sx;         // physical col (group-aligned)
      x[c2] = *(const float4*)&sc[r * S_ + col];
      mx = fmaxf(mx, fmaxf(fmaxf(x[c2].x, x[c2].y), fmaxf(x[c2].z, x[c2].w)));
    }
#pragma unroll
    for (int off = 16; off; off >>= 1)
      mx = fmaxf(mx, __shfl_xor(mx, off, 32));

    float sm = 0.f;
#pragma unroll
    for (int c2 = 0; c2 < 8; ++c2) {
      x[c2].x = __expf(x[c2].x - mx);
      x[c2].y = __expf(x[c2].y - mx);
      x[c2].z = __expf(x[c2].z - mx);
      x[c2].w = __expf(x[c2].w - mx);
      sm += x[c2].x + x[c2].y + x[c2].z + x[c2].w;
    }
#pragma unroll
    for (int off = 16; off; off >>= 1)
      sm += __shfl_xor(sm, off, 32);

    const float inv = __frcp_rn(sm);
    const size_t growBase = (size_t)(rb * 16 + r) * S_;
#pragma unroll
    for (int c2 = 0; c2 < 8; ++c2) {
      const int j   = lane + c2 * 32;
      const int col = (4 * j) ^ sx;
      x[c2].x *= inv; x[c2].y *= inv; x[c2].z *= inv; x[c2].w *= inv;
      *(float4*)&sc[r * S_ + col] = x[c2];          // normalized, back to LDS
      *(float4*)&attnB[growBase + 4 * j] = x[c2];   // coalesced HBM write
    }
  }
  __syncthreads();

  // ---------------- Phase 3: out = scores @ V (split-K) ----------------
  // wave -> (n-tile = wave&3, K-half = wave>>2); partials reduced via LDS.
  const int nt = wave & 3;
  const int kh = wave >> 2;
  const float* __restrict__ vb = v + (size_t)bh * S_ * D_;
  const int col = nt * 16 + m;

  v8f acc = v8f{};
#pragma unroll 4
  for (int kk = kh * 512; kk < kh * 512 + 512; kk += 4) {
    const int ko = kk + half * 2;
    v2f a = *(const v2f*)&sc[swz(m, ko)];      // conflict-free (swizzled)
    v2f bf;
    bf.x = vb[(ko    ) * D_ + col];
    bf.y = vb[(ko + 1) * D_ + col];
    acc = WMMA_F32_16x16x4(a, bf, acc);
  }
  __syncthreads();                 // all LDS score reads complete

  if (kh == 1) {                   // stash partial (reuse score buffer)
#pragma unroll
    for (int r = 0; r < 8; ++r) {
      const int mm = r + half * 8;
      sc[nt * 256 + mm * 16 + m] = acc[r];
    }
  }
  __syncthreads();

  if (kh == 0) {                   // combine + store output
    float* __restrict__ ob = out + (size_t)bh * S_ * D_;
#pragma unroll
    for (int r = 0; r < 8; ++r) {
      const int mm = r + half * 8;
      ob[(rb * 16 + mm) * D_ + col] = acc[r] + sc[nt * 256 + mm * 16 + m];
    }
  }
}

// ---------------------------------------------------------------------------
extern "C" void kernel_launch(void* const* d_in, const int* in_sizes, int n_in,
                              void* d_out, int out_size, void* d_ws, size_t ws_size,
                              hipStream_t stream) {
  const float* q     = (const float*)d_in[0];
  const float* k     = (const float*)d_in[1];
  const float* v     = (const float*)d_in[2];
  const float* mask  = (const float*)d_in[3];
  const int*   amask = (const int*)  d_in[4];
  const float* gamma = (const float*)d_in[5];

  float* out  = (float*)d_out;                              // [B,H,S,D]
  float* attn = (float*)d_out + (size_t)B_ * H_ * S_ * D_;  // [B,H,S,S]

  fused_attention_kernel<<<B_ * H_ * (S_ / 16), 256, 0, stream>>>(
      q, k, v, mask, amask, gamma, out, attn);
}